// RNNDecoder_30313879176079
// MI455X (gfx1250) — compile-verified
//
#include <hip/hip_runtime.h>
#include <hip/hip_bf16.h>
#include <math.h>

// Problem constants (from reference)
#define VOCAB 50000
#define EMB   512
#define HID   512
#define SRC   128
#define BB    64

typedef float v2f __attribute__((ext_vector_type(2)));
typedef float v8f __attribute__((ext_vector_type(8)));

// ---------------------------------------------------------------------------
// Generic WMMA GEMM:  C[64, N] = X[64, K] @ W[N, K]^T + bias   (opt. tanh)
//   - 256 threads = 8 waves; each wave owns a 16-column strip (block: 128 cols)
//   - X staged in dynamic LDS via CDNA5 async global->LDS copies (ASYNCcnt),
//     rows padded by +4 floats (bank-conflict free ds_load fragments)
//   - fp32 matrix core: v_wmma_f32_16x16x4_f32, 4 M-tiles per wave (M=64)
//   - B fragments software-pipelined one k-chunk ahead; W rows prefetched
// ---------------------------------------------------------------------------
__launch_bounds__(256)
__global__ void gemm_xwt_wmma(const float* __restrict__ X,    // [64, K]
                              const float* __restrict__ W,    // [N, K]
                              const float* __restrict__ bias, // [N] or nullptr
                              float* __restrict__ C,          // [64, N]
                              int K, int N, int act)
{
    extern __shared__ float ldsX[];          // 64 * (K+4) floats
    const int KP   = K + 4;                  // padded row stride
    const int tid  = threadIdx.x;

    // ---- stage X into LDS with async global->LDS copies (b128 granularity) ----
    {
        const int row4 = K >> 2;             // float4s per row; BB*row4 % 256 == 0
        const float4* X4 = (const float4*)X;
        for (int i = tid; i < BB * row4; i += 256) {
            int r  = i / row4;
            int c4 = i - r * row4;
            // LDS byte offset = low 32 bits of generic shared-space pointer
            unsigned      ldsAddr = (unsigned)(size_t)(const void*)(ldsX + r * KP + c4 * 4);
            const float4* gsrc    = X4 + i;
            asm volatile("global_load_async_to_lds_b128 %0, %1, off"
                         :: "v"(ldsAddr), "v"(gsrc) : "memory");
        }
        asm volatile("s_wait_asynccnt 0x0" ::: "memory");
    }
    __syncthreads();

    const int wave  = tid >> 5;
    const int lane  = tid & 31;
    const int n0    = blockIdx.x * 128 + wave * 16;
    const int nc    = n0 + (lane & 15);              // this lane's output column
    const int ncl   = (nc < N) ? nc : (N - 1);       // clamped for loads
    const int khalf = (lane >> 4) << 1;              // 0 (lanes 0-15) / 2 (16-31)

    v8f acc[4];
    acc[0] = v8f{}; acc[1] = v8f{}; acc[2] = v8f{}; acc[3] = v8f{};

    const float* wrow  = W + (size_t)ncl * K + khalf;       // B fragment source
    const int    mbase = (lane & 15) * KP + khalf;          // A fragment base

    // B 4x16 fragment: VGPR0=K{0|2}, VGPR1=K{1|3}, N striped across lane%16.
    // Software pipeline: bfrag always holds the *next* chunk's B fragment.
    v2f bfrag = *(const v2f*)(wrow);

    // K is always a multiple of 32 here (512 / 1024).
    for (int k0 = 0; k0 < K; k0 += 32) {
        // one 128B cacheline of this lane's W row per outer step -> prefetch ~1KB ahead
        __builtin_prefetch(wrow + k0 + 256, 0, 3);
        #pragma unroll
        for (int kk = 0; kk < 32; kk += 4) {
            const int k = k0 + kk;
            const v2f bcur = bfrag;

            // Batch all 4 A fragments (keeps them simultaneously live ->
            // distinct VGPRs, clause of ds_loads with staggered waits)
            v2f af[4];
            #pragma unroll
            for (int mt = 0; mt < 4; ++mt) {
                // A 16x4 fragment, M-tile mt: rows m=mt*16+lane%16, cols k+khalf+{0,1}
                af[mt] = *(const v2f*)(ldsX + mbase + mt * 16 * KP + k);
            }

            // Issue next chunk's B load before consuming bcur (hides HBM latency
            // behind the 4 WMMAs below). Last chunk wraps to 0 (dead load, no OOB).
            const int kn = (k + 4 < K) ? (k + 4) : 0;
            bfrag = *(const v2f*)(wrow + kn);

            #pragma unroll
            for (int mt = 0; mt < 4; ++mt) {
                acc[mt] = __builtin_amdgcn_wmma_f32_16x16x4_f32(
                    false, af[mt], false, bcur, (short)0, acc[mt], false, false);
            }
        }
    }

    // ---- epilogue: bias, optional tanh, store (uniform per wave: N % 16 == 0) ----
    if (nc < N) {
        const float bv     = bias ? bias[nc] : 0.0f;
        const int   rowoff = (lane >> 4) * 8;   // C/D: VGPR j -> M=j (lo half), M=8+j (hi)
        #pragma unroll
        for (int mt = 0; mt < 4; ++mt) {
            #pragma unroll
            for (int j = 0; j < 8; ++j) {
                float v = acc[mt][j] + bv;
                if (act) v = tanhf(v);
                C[(size_t)(mt * 16 + rowoff + j) * N + nc] = v;
            }
        }
    }
}

// ---------------------------------------------------------------------------
// Attention: scores = enc_out[b] @ q[b]; masked softmax; ctx = attn @ enc_out[b]
// One block of 128 threads per batch row.
// ---------------------------------------------------------------------------
__launch_bounds__(128)
__global__ void attn_kernel(const float* __restrict__ q,        // [64, HID]
                            const float* __restrict__ enc_out,  // [64, SRC, HID]
                            const unsigned char* __restrict__ mask, // [64, SRC] (bool)
                            float* __restrict__ attn_out,       // [64, SRC]
                            float* __restrict__ ctx)            // [64, HID]
{
    __shared__ float sq[HID];
    __shared__ float sattn[SRC];
    __shared__ float red[SRC];

    const int b = blockIdx.x;
    const int t = threadIdx.x;

    for (int i = t; i < HID; i += 128) sq[i] = q[b * HID + i];
    __syncthreads();

    const float* eb = enc_out + (size_t)b * SRC * HID;

    // score for s = t
    float sc = 0.0f;
    {
        const float* er = eb + t * HID;
        #pragma unroll 4
        for (int h = 0; h < HID; h += 4) {
            float4 e  = *(const float4*)(er + h);
            float4 qq = *(const float4*)(sq + h);
            sc += e.x * qq.x + e.y * qq.y + e.z * qq.z + e.w * qq.w;
        }
    }
    if (!mask[b * SRC + t]) sc = -INFINITY;

    // block-wide max
    red[t] = sc; __syncthreads();
    for (int off = 64; off > 0; off >>= 1) {
        if (t < off) red[t] = fmaxf(red[t], red[t + off]);
        __syncthreads();
    }
    const float mx = red[0]; __syncthreads();

    float e = __expf(sc - mx);
    red[t] = e; __syncthreads();
    for (int off = 64; off > 0; off >>= 1) {
        if (t < off) red[t] += red[t + off];
        __syncthreads();
    }
    const float inv = 1.0f / red[0];

    const float a = e * inv;
    attn_out[b * SRC + t] = a;
    sattn[t] = a;
    __syncthreads();

    // ctx: thread t owns h = 4t..4t+3 (coalesced float4 across threads)
    float cx = 0.f, cy = 0.f, cz = 0.f, cw = 0.f;
    for (int s = 0; s < SRC; ++s) {
        const float  av = sattn[s];
        const float4 e4 = *(const float4*)(eb + s * HID + t * 4);
        cx += av * e4.x; cy += av * e4.y; cz += av * e4.z; cw += av * e4.w;
    }
    float4 out = make_float4(cx, cy, cz, cw);
    *(float4*)(ctx + b * HID + t * 4) = out;
}

// ---------------------------------------------------------------------------
// x0 = concat(emb_weight[ids], ctx)  -> [64, EMB+HID]
// ---------------------------------------------------------------------------
__global__ void build_x0_kernel(const int* __restrict__ ids,
                                const float* __restrict__ embW,
                                const float* __restrict__ ctx,
                                float* __restrict__ x0)
{
    int i = blockIdx.x * 256 + threadIdx.x;       // 64 * 1024
    if (i >= BB * (EMB + HID)) return;
    int b = i >> 10, c = i & 1023;
    x0[i] = (c < EMB) ? embW[(size_t)ids[b] * EMB + c]
                      : ctx[b * HID + (c - EMB)];
}

// out = concat(a, c) -> [64, 1024]
__global__ void concat2_kernel(const float* __restrict__ a,
                               const float* __restrict__ c,
                               float* __restrict__ out)
{
    int i = blockIdx.x * 256 + threadIdx.x;       // 64 * 1024
    if (i >= BB * (HID + HID)) return;
    int b = i >> 10, j = i & 1023;
    out[i] = (j < HID) ? a[b * HID + j] : c[b * HID + (j - HID)];
}

// ---------------------------------------------------------------------------
// GRU gate math (PyTorch order r,z,n):  h = (1-z)*n + z*h_prev
// ---------------------------------------------------------------------------
__global__ void gru_gate_kernel(const float* __restrict__ gi,    // [64, 3*HID]
                                const float* __restrict__ gh,    // [64, 3*HID]
                                const float* __restrict__ hprev, // [64, HID]
                                float* __restrict__ hout_ws,     // [64, HID]
                                float* __restrict__ hout_state)  // [64, HID]
{
    int i = blockIdx.x * 256 + threadIdx.x;       // 64 * 512
    if (i >= BB * HID) return;
    int b = i >> 9, j = i & 511;
    const float* gib = gi + (size_t)b * 3 * HID;
    const float* ghb = gh + (size_t)b * 3 * HID;
    float r = 1.0f / (1.0f + __expf(-(gib[j]           + ghb[j])));
    float z = 1.0f / (1.0f + __expf(-(gib[HID + j]     + ghb[HID + j])));
    float n = tanhf(gib[2 * HID + j] + r * ghb[2 * HID + j]);
    float h = (1.0f - z) * n + z * hprev[i];
    hout_ws[i]    = h;
    hout_state[i] = h;
}

// ---------------------------------------------------------------------------
extern "C" void kernel_launch(void* const* d_in, const int* in_sizes, int n_in,
                              void* d_out, int out_size, void* d_ws, size_t ws_size,
                              hipStream_t stream)
{
    const int*           ids      = (const int*)d_in[0];
    const float*         state    = (const float*)d_in[1];   // [2,64,512]
    const float*         enc_out  = (const float*)d_in[2];   // [64,128,512]
    const unsigned char* enc_mask = (const unsigned char*)d_in[3]; // bool [64,128]
    const float*         embW     = (const float*)d_in[4];   // [50000,512]
    const float*         attn_W   = (const float*)d_in[5];   // [512,512]
    const float*         W_ih0    = (const float*)d_in[6];   // [1536,1024]
    const float*         W_hh0    = (const float*)d_in[7];   // [1536,512]
    const float*         b_ih0    = (const float*)d_in[8];
    const float*         b_hh0    = (const float*)d_in[9];
    const float*         W_ih1    = (const float*)d_in[10];  // [1536,512]
    const float*         W_hh1    = (const float*)d_in[11];  // [1536,512]
    const float*         b_ih1    = (const float*)d_in[12];
    const float*         b_hh1    = (const float*)d_in[13];
    const float*         pre_W    = (const float*)d_in[14];  // [512,1024]
    const float*         pre_b    = (const float*)d_in[15];
    const float*         out_bias = (const float*)d_in[16];

    const float* state0 = state;
    const float* state1 = state + BB * HID;

    // d_out layout: logits | new_state | attn
    float* out_logits = (float*)d_out;                       // 64*50000
    float* out_state0 = out_logits + (size_t)BB * VOCAB;     // 64*512
    float* out_state1 = out_state0 + BB * HID;               // 64*512
    float* out_attn   = out_state1 + BB * HID;               // 64*128

    // workspace layout (floats)
    float* ws  = (float*)d_ws;
    float* q   = ws;                      // 32768 (reused later for 'pre')
    float* ctx = q   + BB * HID;          // 32768
    float* x0  = ctx + BB * HID;          // 65536 (reused later for pre-input concat)
    float* gi  = x0  + BB * (EMB + HID);  // 98304
    float* gh  = gi  + BB * 3 * HID;      // 98304
    float* h0  = gh  + BB * 3 * HID;      // 32768
    float* h1  = h0  + BB * HID;          // 32768
    float* pre = q;                       // reuse
    float* xp  = x0;                      // reuse

    const dim3 tb(256);
    auto ldsBytes = [](int K) { return (size_t)BB * (K + 4) * sizeof(float); };

    // 1) q = state1 @ attn_W^T                         [64,512]
    gemm_xwt_wmma<<<dim3((HID + 127) / 128), tb, ldsBytes(HID), stream>>>(
        state1, attn_W, nullptr, q, HID, HID, 0);

    // 2) attention: scores/softmax/ctx (+ attn output)
    attn_kernel<<<dim3(BB), dim3(128), 0, stream>>>(q, enc_out, enc_mask, out_attn, ctx);

    // 3) x0 = [emb, ctx]
    build_x0_kernel<<<dim3(BB * (EMB + HID) / 256), tb, 0, stream>>>(ids, embW, ctx, x0);

    // 4) GRU layer 0
    gemm_xwt_wmma<<<dim3((3 * HID + 127) / 128), tb, ldsBytes(EMB + HID), stream>>>(
        x0, W_ih0, b_ih0, gi, EMB + HID, 3 * HID, 0);
    gemm_xwt_wmma<<<dim3((3 * HID + 127) / 128), tb, ldsBytes(HID), stream>>>(
        state0, W_hh0, b_hh0, gh, HID, 3 * HID, 0);
    gru_gate_kernel<<<dim3(BB * HID / 256), tb, 0, stream>>>(gi, gh, state0, h0, out_state0);

    // 5) GRU layer 1
    gemm_xwt_wmma<<<dim3((3 * HID + 127) / 128), tb, ldsBytes(HID), stream>>>(
        h0, W_ih1, b_ih1, gi, HID, 3 * HID, 0);
    gemm_xwt_wmma<<<dim3((3 * HID + 127) / 128), tb, ldsBytes(HID), stream>>>(
        state1, W_hh1, b_hh1, gh, HID, 3 * HID, 0);
    gru_gate_kernel<<<dim3(BB * HID / 256), tb, 0, stream>>>(gi, gh, state1, h1, out_state1);

    // 6) pre = tanh([h1, ctx] @ pre_W^T + pre_b)       [64,512]
    concat2_kernel<<<dim3(BB * (HID + HID) / 256), tb, 0, stream>>>(h1, ctx, xp);
    gemm_xwt_wmma<<<dim3((EMB + 127) / 128), tb, ldsBytes(2 * HID), stream>>>(
        xp, pre_W, pre_b, pre, 2 * HID, EMB, 1 /*tanh*/);

    // 7) logits = pre @ emb_weight^T + out_bias        [64,50000]  (the 102 MB GEMM)
    gemm_xwt_wmma<<<dim3((VOCAB + 127) / 128), tb, ldsBytes(EMB), stream>>>(
        pre, embW, out_bias, out_logits, EMB, VOCAB, 0);
}